// GNNModel_13769665151624
// MI455X (gfx1250) — compile-verified
//
#include <hip/hip_runtime.h>

#define BN_EPS 1e-5f

typedef __attribute__((ext_vector_type(2))) float v2f;
typedef __attribute__((ext_vector_type(8))) float v8f;

// ---------------------------------------------------------------------------
// Tiled GEMM on the CDNA5 tensor path: C[M x Nc] = A[M x K] @ W[K x Nc]
// One wave (32 lanes) per 16x64 output strip: four independent 16x16
// accumulator chains (hides V_WMMA_F32_16X16X4_F32 pipeline latency) and the
// A fragment is loaded once per k-step and shared by all four WMMAs.
// Requires: M % 16 == 0, Nc % 64 == 0 (true for all call sites here).
// Fragment layouts per CDNA5 ISA 7.12.2:
//   A 16x4:  lanes 0-15 hold K=0,1 ; lanes 16-31 hold K=2,3 ; M = lane&15
//   B 4x16:  mirrored (N = lane&15, K pair selected by lane half)
//   C/D:     VGPR r -> row r (lanes 0-15) / row r+8 (lanes 16-31), N = lane&15
// ---------------------------------------------------------------------------
__global__ __launch_bounds__(32) void gemm_wmma_kernel(
    const float* __restrict__ A, const float* __restrict__ W,
    const float* __restrict__ bias, float* __restrict__ C,
    int K, int Nc, int applyRelu)
{
    const int colBase = blockIdx.x * 64;
    const int rowBase = blockIdx.y * 16;
    const int lane = threadIdx.x;
    const int half = lane >> 4;
    const int l16  = lane & 15;

    const float* __restrict__ Arow = A + (size_t)(rowBase + l16) * K;
    const float* __restrict__ Wc0 = W + colBase + l16;
    const float* __restrict__ Wc1 = Wc0 + 16;
    const float* __restrict__ Wc2 = Wc0 + 32;
    const float* __restrict__ Wc3 = Wc0 + 48;

    v8f c0 = {}, c1 = {}, c2 = {}, c3 = {};
    for (int k0 = 0; k0 < K; k0 += 4) {
        const int ka = k0 + 2 * half;
        const size_t r0 = (size_t)ka * Nc;
        const size_t r1 = (size_t)(ka + 1) * Nc;
        v2f a = *(const v2f*)(Arow + ka);            // 8B aligned: ka even
        v2f b0; b0.x = Wc0[r0]; b0.y = Wc0[r1];
        v2f b1; b1.x = Wc1[r0]; b1.y = Wc1[r1];
        v2f b2; b2.x = Wc2[r0]; b2.y = Wc2[r1];
        v2f b3; b3.x = Wc3[r0]; b3.y = Wc3[r1];
        c0 = __builtin_amdgcn_wmma_f32_16x16x4_f32(false, a, false, b0, (short)0, c0, false, false);
        c1 = __builtin_amdgcn_wmma_f32_16x16x4_f32(false, a, false, b1, (short)0, c1, false, false);
        c2 = __builtin_amdgcn_wmma_f32_16x16x4_f32(false, a, false, b2, (short)0, c2, false, false);
        c3 = __builtin_amdgcn_wmma_f32_16x16x4_f32(false, a, false, b3, (short)0, c3, false, false);
    }

#pragma unroll
    for (int t = 0; t < 4; ++t) {
        const v8f& c = (t == 0) ? c0 : (t == 1) ? c1 : (t == 2) ? c2 : c3;
        const int col = colBase + 16 * t + l16;
        const float bv = bias ? bias[col] : 0.0f;
#pragma unroll
        for (int r = 0; r < 8; ++r) {
            const int row = rowBase + r + 8 * half;
            float v = c[r] + bv;
            if (applyRelu) v = fmaxf(v, 0.0f);
            C[(size_t)row * Nc + col] = v;
        }
    }
}

// ---------------------------------------------------------------------------
// Degree / normalization helpers
// ---------------------------------------------------------------------------
__global__ void fill_kernel(float* __restrict__ p, float v, int n) {
    int i = blockIdx.x * blockDim.x + threadIdx.x;
    if (i < n) p[i] = v;
}

__global__ void deg_accum_kernel(const long long* __restrict__ dst,
                                 float* __restrict__ deg, int E_) {
    int e = blockIdx.x * blockDim.x + threadIdx.x;
    if (e < E_) atomicAdd(&deg[(int)dst[e]], 1.0f);
}

__global__ void rsqrt_kernel(float* __restrict__ p, int n) {
    int i = blockIdx.x * blockDim.x + threadIdx.x;
    if (i < n) p[i] = rsqrtf(p[i]);
}

// ---------------------------------------------------------------------------
// Edge aggregation: agg[dst] += h[src] * dinv[src]*dinv[dst]
// One 32-lane wave covers one edge's feature row (float4 per lane):
// coalesced gather of the L2-resident h[src] row, float atomics into agg[dst].
// ---------------------------------------------------------------------------
__global__ void edge_agg_kernel(const long long* __restrict__ src,
                                const long long* __restrict__ dst,
                                const float* __restrict__ dinv,
                                const float* __restrict__ h,
                                float* __restrict__ agg,
                                int E_, int F) {
    const int chunks = F >> 2;
    int i = blockIdx.x * blockDim.x + threadIdx.x;
    int total = E_ * chunks;
    if (i >= total) return;
    int e = i / chunks;
    int c = (i - e * chunks) << 2;
    int s = (int)src[e];
    int d = (int)dst[e];
    float norm = dinv[s] * dinv[d];
    const float4 v = *(const float4*)(h + (size_t)s * F + c);
    float* o = agg + (size_t)d * F + c;
    atomicAdd(o + 0, v.x * norm);
    atomicAdd(o + 1, v.y * norm);
    atomicAdd(o + 2, v.z * norm);
    atomicAdd(o + 3, v.w * norm);
}

// ---------------------------------------------------------------------------
// Fused epilogue: out = [BN]( agg + h*dinv^2 + b ) with optional ReLU
// ---------------------------------------------------------------------------
__global__ void post_kernel(const float* __restrict__ agg,
                            const float* __restrict__ h,
                            const float* __restrict__ dinv,
                            const float* __restrict__ b,
                            const float* __restrict__ g,
                            const float* __restrict__ beta,
                            const float* __restrict__ mean,
                            const float* __restrict__ var,
                            float* __restrict__ out,
                            int Nn, int F, int doBN, int doRelu) {
    int i = blockIdx.x * blockDim.x + threadIdx.x;
    int total = Nn * F;
    if (i >= total) return;
    int n = i / F;
    int f = i - n * F;
    float di = dinv[n];
    float t = agg[i] + h[i] * di * di + b[f];
    if (doBN) t = (t - mean[f]) * rsqrtf(var[f] + BN_EPS) * g[f] + beta[f];
    if (doRelu) t = fmaxf(t, 0.0f);
    out[i] = t;
}

// ---------------------------------------------------------------------------
// gate[n] = dot(gh[n,:], Wg2) + bg2
// ---------------------------------------------------------------------------
__global__ void gate_kernel(const float* __restrict__ gh,
                            const float* __restrict__ Wg2,
                            const float* __restrict__ bg2,
                            float* __restrict__ gate, int Nn, int F) {
    int n = blockIdx.x * blockDim.x + threadIdx.x;
    if (n >= Nn) return;
    const float* row = gh + (size_t)n * F;
    float acc = bg2[0];
    for (int f = 0; f < F; f += 4) {
        float4 v = *(const float4*)(row + f);
        float4 w = *(const float4*)(Wg2 + f);
        acc += v.x * w.x + v.y * w.y + v.z * w.z + v.w * w.w;
    }
    gate[n] = acc;
}

// ---------------------------------------------------------------------------
// Segment softmax (order-preserving uint encoding for float atomicMax)
// ---------------------------------------------------------------------------
__device__ __forceinline__ unsigned encf(float x) {
    unsigned u = __float_as_uint(x);
    return (u & 0x80000000u) ? ~u : (u | 0x80000000u);
}
__device__ __forceinline__ float decf(unsigned u) {
    u = (u & 0x80000000u) ? (u & 0x7FFFFFFFu) : ~u;
    return __uint_as_float(u);
}

__global__ void seg_max_kernel(const float* __restrict__ gate,
                               const long long* __restrict__ batch,
                               unsigned* __restrict__ mEnc, int Nn) {
    int n = blockIdx.x * blockDim.x + threadIdx.x;
    if (n >= Nn) return;
    atomicMax(&mEnc[(int)batch[n]], encf(gate[n]));
}

__global__ void seg_expsum_kernel(const float* __restrict__ gate,
                                  const long long* __restrict__ batch,
                                  const unsigned* __restrict__ mEnc,
                                  float* __restrict__ eBuf,
                                  float* __restrict__ sBuf, int Nn) {
    int n = blockIdx.x * blockDim.x + threadIdx.x;
    if (n >= Nn) return;
    int g = (int)batch[n];
    float ev = expf(gate[n] - decf(mEnc[g]));
    eBuf[n] = ev;
    atomicAdd(&sBuf[g], ev);
}

// ---------------------------------------------------------------------------
// pooled[g,:] += h3[n,:] * (e[n]/s[g]) with LDS staging.
// batch is sorted, so each 256-node block spans very few segments: accumulate
// into a 64x64 LDS tile (LDS atomics), then flush only rows [gmin,gmax].
// ---------------------------------------------------------------------------
__global__ void pooled_kernel(const float* __restrict__ h3,
                              const float* __restrict__ eBuf,
                              const float* __restrict__ sBuf,
                              const long long* __restrict__ batch,
                              float* __restrict__ pooled, int Nn) {
    __shared__ float tile[64 * 64];
    __shared__ int gminS, gmaxS;
    int n = blockIdx.x * blockDim.x + threadIdx.x;
    if (threadIdx.x == 0) { gminS = 63; gmaxS = 0; }
    __syncthreads();
    int g = -1;
    if (n < Nn) {
        g = (int)batch[n];
        atomicMin(&gminS, g);
        atomicMax(&gmaxS, g);
    }
    __syncthreads();
    int gmin = gminS, gmax = gmaxS;
    if (gmax >= gmin) {
        int nent = (gmax - gmin + 1) * 64;
        for (int i = threadIdx.x; i < nent; i += blockDim.x)
            tile[gmin * 64 + i] = 0.0f;
    }
    __syncthreads();
    if (n < Nn) {
        float a = eBuf[n] / sBuf[g];
        const float* row = h3 + (size_t)n * 64;
        for (int f = 0; f < 64; ++f)
            atomicAdd(&tile[g * 64 + f], row[f] * a);
    }
    __syncthreads();
    if (gmax >= gmin) {
        int nent = (gmax - gmin + 1) * 64;
        for (int i = threadIdx.x; i < nent; i += blockDim.x)
            atomicAdd(&pooled[gmin * 64 + i], tile[gmin * 64 + i]);
    }
}

// ---------------------------------------------------------------------------
// Host-side orchestration
// ---------------------------------------------------------------------------
extern "C" void kernel_launch(void* const* d_in, const int* in_sizes, int n_in,
                              void* d_out, int out_size, void* d_ws, size_t ws_size,
                              hipStream_t stream) {
    const int DIN = 128, DH = 128, DOUT = 64, G = 64;

    const float*     x     = (const float*)d_in[0];
    const long long* edge  = (const long long*)d_in[1];
    const long long* batch = (const long long*)d_in[2];
    const float* W1 = (const float*)d_in[3];  const float* b1 = (const float*)d_in[4];
    const float* W2 = (const float*)d_in[5];  const float* b2 = (const float*)d_in[6];
    const float* W3 = (const float*)d_in[7];  const float* b3 = (const float*)d_in[8];
    const float* g1 = (const float*)d_in[9];  const float* be1 = (const float*)d_in[10];
    const float* g2 = (const float*)d_in[11]; const float* be2 = (const float*)d_in[12];
    const float* mean1 = (const float*)d_in[13]; const float* var1 = (const float*)d_in[14];
    const float* mean2 = (const float*)d_in[15]; const float* var2 = (const float*)d_in[16];
    const float* Wg1 = (const float*)d_in[17]; const float* bg1 = (const float*)d_in[18];
    const float* Wg2 = (const float*)d_in[19]; const float* bg2 = (const float*)d_in[20];
    const float* Wm1 = (const float*)d_in[21]; const float* bm1 = (const float*)d_in[22];
    const float* Wm2 = (const float*)d_in[23]; const float* bm2 = (const float*)d_in[24];

    const int N = in_sizes[0] / DIN;   // 100000 (multiple of 16)
    const int E = in_sizes[1] / 2;     // 1000000
    const long long* srcIdx = edge;
    const long long* dstIdx = edge + E;

    // Workspace carving (256B aligned)
    char* ws = (char*)d_ws;
    size_t off = 0;
    auto carve = [&](size_t bytes) -> void* {
        void* p = ws + off;
        off = (off + bytes + 255) & ~(size_t)255;
        return p;
    };
    float*    dinv   = (float*)carve((size_t)N * 4);
    float*    buf0   = (float*)carve((size_t)N * DH * 4);   // h = A@W
    float*    buf1   = (float*)carve((size_t)N * DH * 4);   // agg / layer out
    float*    buf2   = (float*)carve((size_t)N * DH * 4);   // agg / layer out
    float*    gateB  = (float*)carve((size_t)N * 4);
    float*    eBuf   = (float*)carve((size_t)N * 4);
    unsigned* mEnc   = (unsigned*)carve((size_t)G * 4);
    float*    sBuf   = (float*)carve((size_t)G * 4);
    float*    pooled = (float*)carve((size_t)G * DOUT * 4);
    float*    t1     = (float*)carve((size_t)G * DH * 4);
    (void)ws_size; (void)n_in; (void)out_size;

    const dim3 B256(256);
    const dim3 WAVE(32);
    const int nbN = (N + 255) / 256;
    const int nbE = (E + 255) / 256;

    // degree -> dinv = rsqrt(1 + indegree)
    fill_kernel<<<nbN, B256, 0, stream>>>(dinv, 1.0f, N);
    deg_accum_kernel<<<nbE, B256, 0, stream>>>(dstIdx, dinv, E);
    rsqrt_kernel<<<nbN, B256, 0, stream>>>(dinv, N);

    // ---- Layer 1: h = x@W1 ; agg ; out = relu(bn(agg + h*dinv^2 + b1))
    gemm_wmma_kernel<<<dim3(DH / 64, N / 16), WAVE, 0, stream>>>(x, W1, nullptr, buf0, DIN, DH, 0);
    hipMemsetAsync(buf1, 0, (size_t)N * DH * 4, stream);
    {
        int total = E * (DH / 4);
        edge_agg_kernel<<<(total + 255) / 256, B256, 0, stream>>>(srcIdx, dstIdx, dinv, buf0, buf1, E, DH);
    }
    post_kernel<<<(N * DH + 255) / 256, B256, 0, stream>>>(buf1, buf0, dinv, b1, g1, be1, mean1, var1,
                                                           buf1, N, DH, 1, 1);

    // ---- Layer 2
    gemm_wmma_kernel<<<dim3(DH / 64, N / 16), WAVE, 0, stream>>>(buf1, W2, nullptr, buf0, DH, DH, 0);
    hipMemsetAsync(buf2, 0, (size_t)N * DH * 4, stream);
    {
        int total = E * (DH / 4);
        edge_agg_kernel<<<(total + 255) / 256, B256, 0, stream>>>(srcIdx, dstIdx, dinv, buf0, buf2, E, DH);
    }
    post_kernel<<<(N * DH + 255) / 256, B256, 0, stream>>>(buf2, buf0, dinv, b2, g2, be2, mean2, var2,
                                                           buf2, N, DH, 1, 1);

    // ---- Layer 3 (no BN / no ReLU): h3 -> buf1 (N x 64)
    gemm_wmma_kernel<<<dim3(DOUT / 64, N / 16), WAVE, 0, stream>>>(buf2, W3, nullptr, buf0, DH, DOUT, 0);
    hipMemsetAsync(buf1, 0, (size_t)N * DOUT * 4, stream);
    {
        int total = E * (DOUT / 4);
        edge_agg_kernel<<<(total + 255) / 256, B256, 0, stream>>>(srcIdx, dstIdx, dinv, buf0, buf1, E, DOUT);
    }
    post_kernel<<<(N * DOUT + 255) / 256, B256, 0, stream>>>(buf1, buf0, dinv, b3,
                                                             nullptr, nullptr, nullptr, nullptr,
                                                             buf1, N, DOUT, 0, 0);

    // ---- Gate MLP: gh = relu(h3@Wg1 + bg1) ; gate = gh@Wg2 + bg2
    gemm_wmma_kernel<<<dim3(DH / 64, N / 16), WAVE, 0, stream>>>(buf1, Wg1, bg1, buf0, DOUT, DH, 1);
    gate_kernel<<<nbN, B256, 0, stream>>>(buf0, Wg2, bg2, gateB, N, DH);

    // ---- Segment softmax + attention pooling
    hipMemsetAsync(mEnc, 0, (size_t)G * 4, stream);
    hipMemsetAsync(sBuf, 0, (size_t)G * 4, stream);
    hipMemsetAsync(pooled, 0, (size_t)G * DOUT * 4, stream);
    seg_max_kernel<<<nbN, B256, 0, stream>>>(gateB, batch, mEnc, N);
    seg_expsum_kernel<<<nbN, B256, 0, stream>>>(gateB, batch, mEnc, eBuf, sBuf, N);
    pooled_kernel<<<nbN, B256, 0, stream>>>(buf1, eBuf, sBuf, batch, pooled, N);

    // ---- Readout MLP: out = relu(pooled@Wm1 + bm1) @ Wm2 + bm2
    gemm_wmma_kernel<<<dim3(DH / 64, G / 16), WAVE, 0, stream>>>(pooled, Wm1, bm1, t1, DOUT, DH, 1);
    gemm_wmma_kernel<<<dim3(DOUT / 64, G / 16), WAVE, 0, stream>>>(t1, Wm2, bm2, (float*)d_out, DH, DOUT, 0);
}